// SaltAndPepper_72834055405766
// MI455X (gfx1250) — compile-verified
//
#include <hip/hip_runtime.h>
#include <hip/hip_bf16.h>

// SaltAndPepper blend: out = x*a + (1-a)*b  ==  fma(a, x-b, b)
// Pure streaming kernel: 16 B moved per element, 2 FLOPs per element.
// Bandwidth-bound at 23.3 TB/s -> ~92 us floor for 2 GiB of traffic.
// Strategy: B128 (float4) non-temporal loads/stores (gfx1250 TH=NT cache
// policy) so the 4 x 512 MB streams bypass/flow through L2 without thrashing,
// one v_sub + one v_fma per lane element.

typedef float f32x4 __attribute__((ext_vector_type(4)));

__global__ __launch_bounds__(256) void
saltpepper_stream_kernel(const float* __restrict__ x,
                         const float* __restrict__ a,
                         const float* __restrict__ b,
                         float* __restrict__ out,
                         long long n4,      // number of float4 packets
                         long long n_tail,  // leftover scalars (n % 4)
                         long long n)       // total scalar elements
{
    const long long tid = (long long)blockIdx.x * blockDim.x + threadIdx.x;

    if (tid < n4) {
        const f32x4* __restrict__ x4 = (const f32x4*)x;
        const f32x4* __restrict__ a4 = (const f32x4*)a;
        const f32x4* __restrict__ b4 = (const f32x4*)b;
        f32x4* __restrict__       o4 = (f32x4*)out;

        // Three independent B128 NT loads in flight before any consumption.
        f32x4 xv = __builtin_nontemporal_load(x4 + tid);
        f32x4 av = __builtin_nontemporal_load(a4 + tid);
        f32x4 bv = __builtin_nontemporal_load(b4 + tid);

        f32x4 r;
#pragma unroll
        for (int k = 0; k < 4; ++k) {
            // a*(x-b)+b == x*a + (1-a)*b ; exact for a in {0,1}
            r[k] = __builtin_fmaf(av[k], xv[k] - bv[k], bv[k]);
        }
        __builtin_nontemporal_store(r, o4 + tid);
    }

    // Scalar tail (n % 4 leftovers) handled by the first few threads.
    if (tid < n_tail) {
        const long long i = n4 * 4 + tid;
        if (i < n) {
            const float xs = x[i];
            const float as = a[i];
            const float bs = b[i];
            out[i] = __builtin_fmaf(as, xs - bs, bs);
        }
    }
}

extern "C" void kernel_launch(void* const* d_in, const int* in_sizes, int n_in,
                              void* d_out, int out_size, void* d_ws, size_t ws_size,
                              hipStream_t stream) {
    (void)n_in; (void)d_ws; (void)ws_size;

    const float* x = (const float*)d_in[0];
    const float* a = (const float*)d_in[1];
    const float* b = (const float*)d_in[2];
    float* out     = (float*)d_out;

    const long long n      = (long long)out_size;   // 32*2048*2048 = 134,217,728
    const long long n4     = n >> 2;                // float4 packets
    const long long n_tail = n & 3;

    const int  threads = 256;                       // 8 wave32s per block
    long long  work    = (n4 > n_tail) ? n4 : n_tail;
    long long  blocks  = (work + threads - 1) / threads;
    if (blocks < 1) blocks = 1;

    saltpepper_stream_kernel<<<(unsigned)blocks, threads, 0, stream>>>(
        x, a, b, out, n4, n_tail, n);
}